// Upsampling_22204980920920
// MI455X (gfx1250) — compile-verified
//
#include <hip/hip_runtime.h>

typedef float v4f __attribute__((ext_vector_type(4)));

namespace {
constexpr int kB = 32;
constexpr int kS = 512;
constexpr int kD = 384;                 // floats per frame
constexpr int kD4 = kD / 4;             // 96 float4 per frame
constexpr int kThreads2 = 256;          // 8 wave32s
constexpr int kWaves = kThreads2 / 32;
constexpr int kFramesPerWave = 8;
constexpr int kFramesPerBlock = kWaves * kFramesPerWave;  // 64
}

// ---------------------------------------------------------------------------
// Kernel 1: per-row duration round + inclusive scan (LDS) + mel_len
// grid = B blocks, block = S threads (16 wave32s)
// ---------------------------------------------------------------------------
__global__ void __launch_bounds__(kS)
dur_scan_kernel(const float* __restrict__ logd,
                float* __restrict__ dur_out,      // [B,S]  (d_out section)
                float* __restrict__ mellen_out,   // [B]    (d_out section)
                int*   __restrict__ cum_ws,       // [B,S]  workspace
                int*   __restrict__ mellen_ws,    // [B]    workspace
                int max_len)
{
    __shared__ int sc[kS];
    const int b = blockIdx.x;
    const int s = threadIdx.x;

    // duration_rounded = clamp(round(exp(v) - 1), min=0); rintf == round-half-even
    const float v = logd[b * kS + s];
    const float r = fmaxf(rintf(expf(v) - 1.0f), 0.0f);
    dur_out[b * kS + s] = r;

    sc[s] = (int)r;
    __syncthreads();

    // Hillis-Steele inclusive scan over 512 ints
    #pragma unroll
    for (int off = 1; off < kS; off <<= 1) {
        const int add = (s >= off) ? sc[s - off] : 0;
        __syncthreads();
        sc[s] += add;
        __syncthreads();
    }

    cum_ws[b * kS + s] = sc[s];
    if (s == kS - 1) {
        const int ml = min(sc[kS - 1], max_len);
        mellen_ws[b]  = ml;
        mellen_out[b] = (float)ml;
    }
}

// ---------------------------------------------------------------------------
// Kernel 2: gather/expand.  grid = (ceil(T/64), B), block = 256 threads.
// cum row cached in LDS; one wave32 per frame; 3x float4 per lane per frame.
// Streamed output uses non-temporal stores (don't pollute L2; x stays hot).
// ---------------------------------------------------------------------------
__global__ void __launch_bounds__(kThreads2)
gather_kernel(const float* __restrict__ x,        // [B,S,D]
              const int*   __restrict__ cum_ws,   // [B,S]
              const int*   __restrict__ mellen_ws,// [B]
              float* __restrict__ out,            // [B,T,D]
              float* __restrict__ mask_out,       // [B,T]  1.0 = padding
              int T)
{
    __shared__ int sc[kS];
    __shared__ int s_mellen;

    const int b   = blockIdx.y;
    const int tid = threadIdx.x;

    for (int i = tid; i < kS; i += kThreads2) sc[i] = cum_ws[b * kS + i];
    if (tid == 0) s_mellen = mellen_ws[b];
    __syncthreads();

    const int mellen = s_mellen;
    const int wave = tid >> 5;       // wave32
    const int lane = tid & 31;
    const int tile = blockIdx.x * kFramesPerBlock + wave * kFramesPerWave;
    const v4f z = (v4f)0.0f;

    #pragma unroll
    for (int i = 0; i < kFramesPerWave; ++i) {
        const int t = tile + i;
        if (t >= T) break;

        const bool valid = (t < mellen);
        v4f* __restrict__ dst = (v4f*)(out + ((size_t)b * T + t) * kD);

        if (valid) {
            // searchsorted(cum, t, 'right'): first j with cum[j] > t
            int lo = 0, hi = kS;
            #pragma unroll
            for (int step = 0; step < 9; ++step) {
                const int mid = (lo + hi) >> 1;
                if (sc[mid] <= t) lo = mid + 1; else hi = mid;
            }
            const int idx = (lo < kS) ? lo : (kS - 1);
            const v4f* __restrict__ src = (const v4f*)(x + ((size_t)b * kS + idx) * kD);
            #pragma unroll
            for (int k = 0; k < kD4 / 32; ++k) {           // 3 x b128 per lane
                const v4f val = src[lane + 32 * k];
                __builtin_nontemporal_store(val, dst + lane + 32 * k);
            }
        } else {
            #pragma unroll
            for (int k = 0; k < kD4 / 32; ++k)
                __builtin_nontemporal_store(z, dst + lane + 32 * k);
        }
        if (lane == 0)
            mask_out[(size_t)b * T + t] = valid ? 0.0f : 1.0f;
    }
}

// ---------------------------------------------------------------------------
extern "C" void kernel_launch(void* const* d_in, const int* in_sizes, int n_in,
                              void* d_out, int out_size, void* d_ws, size_t ws_size,
                              hipStream_t stream)
{
    const float* x    = (const float*)d_in[0];   // [B,S,D]
    const float* logd = (const float*)d_in[1];   // [B,S]
    (void)in_sizes; (void)n_in; (void)ws_size;

    // out_size = B*T*D + B*S + B + B*T  ->  T = max_len (3072 for this setup)
    const int T = (out_size - kB * kS - kB) / (kB * (kD + 1));

    float* out        = (float*)d_out;
    float* dur_out    = out + (size_t)kB * T * kD;
    float* mellen_out = dur_out + (size_t)kB * kS;
    float* mask_out   = mellen_out + kB;

    int* cum_ws    = (int*)d_ws;            // B*S ints = 64 KB
    int* mellen_ws = cum_ws + kB * kS;      // B ints

    dur_scan_kernel<<<kB, kS, 0, stream>>>(logd, dur_out, mellen_out,
                                           cum_ws, mellen_ws, T);

    dim3 grid((T + kFramesPerBlock - 1) / kFramesPerBlock, kB);
    gather_kernel<<<grid, kThreads2, 0, stream>>>(x, cum_ws, mellen_ws,
                                                  out, mask_out, T);
}